// GCN_46986942218648
// MI455X (gfx1250) — compile-verified
//
#include <hip/hip_runtime.h>

typedef __attribute__((ext_vector_type(2))) float v2f;
typedef __attribute__((ext_vector_type(8))) float v8f;

#define NNODES 20000
#define NEDGES 640000

// ---------------------------------------------------------------- utilities

__global__ void zero_f32_kernel(float* __restrict__ p, int n) {
    int i = blockIdx.x * blockDim.x + threadIdx.x;
    if (i < n) p[i] = 0.0f;
}

// ------------------------------------------------- edge scatter (aggregate)

// 128-channel scatter: one thread handles one (edge, 4-channel group).
// float4 gather from x[src], 4 native f32 atomics into agg[dst]. L2-resident.
__global__ void scatter_add_128(const float* __restrict__ x,
                                const int* __restrict__ src,
                                const int* __restrict__ dst,
                                float* __restrict__ agg) {
    long long t = (long long)blockIdx.x * blockDim.x + threadIdx.x;
    int e = (int)(t >> 5);          // 32 groups of 4 channels per edge
    int g = (int)(t & 31);
    if (e >= NEDGES) return;
    int s = src[e];
    int d = dst[e];
    const float4* xs = (const float4*)(x + (long long)s * 128) + g;
    float4 v = *xs;
    float* ap = agg + (long long)d * 128 + g * 4;
    unsafeAtomicAdd(ap + 0, v.x);
    unsafeAtomicAdd(ap + 1, v.y);
    unsafeAtomicAdd(ap + 2, v.z);
    unsafeAtomicAdd(ap + 3, v.w);
}

// 32-channel scatter: one lane per channel, one wave per edge.
__global__ void scatter_add_32(const float* __restrict__ h,
                               const int* __restrict__ src,
                               const int* __restrict__ dst,
                               float* __restrict__ agg) {
    long long t = (long long)blockIdx.x * blockDim.x + threadIdx.x;
    int e = (int)(t >> 5);
    int c = (int)(t & 31);
    if (e >= NEDGES) return;
    float v = h[(long long)src[e] * 32 + c];
    unsafeAtomicAdd(agg + (long long)dst[e] * 32 + c, v);
}

// ------------------------------------------------------- WMMA f32 16x16x4

// Accumulate a 16x16 output tile of A[m0:m0+16, :K] @ B[:K, n0:n0+16].
// A row-major lda, B row-major ldb. Per-lane fragments follow the CDNA5
// V_WMMA_F32_16X16X4_F32 layouts (A: M=lane%16, K=k0+2*(lane/16)+{0,1};
// B: N=lane%16, same K pair; C/D: M=v+8*(lane/16), N=lane%16).
__device__ __forceinline__ v8f wmma_gemm_tile(const float* __restrict__ A, int lda, int m0,
                                              const float* __restrict__ B, int ldb, int n0,
                                              int K, v8f acc) {
    int lane = threadIdx.x & 31;
    int lm   = lane & 15;
    int hi   = lane >> 4;             // 0 or 1
    #pragma unroll 4
    for (int k0 = 0; k0 < K; k0 += 4) {
        int ka = k0 + hi * 2;
        const float* ap = A + (long long)(m0 + lm) * lda + ka;
        v2f a, b;
        a.x = ap[0];
        a.y = ap[1];
        const float* bp = B + (long long)ka * ldb + (n0 + lm);
        b.x = bp[0];
        b.y = bp[ldb];
        acc = __builtin_amdgcn_wmma_f32_16x16x4_f32(
            /*neg_a=*/false, a, /*neg_b=*/false, b,
            /*c_mod=*/(short)0, acc, /*reuse_a=*/false, /*reuse_b=*/false);
    }
    return acc;
}

__device__ __forceinline__ void wmma_store_tile(float* __restrict__ C, int ldc,
                                                int m0, int n0, v8f acc,
                                                const float* __restrict__ bias,
                                                bool relu) {
    int lane = threadIdx.x & 31;
    int lm   = lane & 15;
    int mo   = (lane >> 4) * 8;
    float bv = bias[n0 + lm];
    #pragma unroll
    for (int v = 0; v < 8; ++v) {
        float val = acc[v] + bv;
        if (relu) val = fmaxf(val, 0.0f);
        C[(long long)(m0 + mo + v) * ldc + n0 + lm] = val;
    }
}

// Layer 1: h1[N,64] = agg1[N,128]@w_rel1 + x[N,128]@w_root1 + b_rel1
// grid.x = 1250 M-tiles, 4 waves/block cover the 4 N-tiles.
__global__ void gcn_layer1_wmma(const float* __restrict__ agg1,
                                const float* __restrict__ x,
                                const float* __restrict__ wrel,
                                const float* __restrict__ wroot,
                                const float* __restrict__ bias,
                                float* __restrict__ h1) {
    int m0 = blockIdx.x * 16;
    int n0 = (threadIdx.x >> 5) * 16;
    v8f acc = {};
    acc = wmma_gemm_tile(agg1, 128, m0, wrel, 64, n0, 128, acc);
    acc = wmma_gemm_tile(x,   128, m0, wroot, 64, n0, 128, acc);
    wmma_store_tile(h1, 64, m0, n0, acc, bias, false);
}

// h[N,32] = relu(h1[N,64] @ w_l1 + b_l1); 2 waves/block (2 N-tiles).
__global__ void linear1_relu_wmma(const float* __restrict__ h1,
                                  const float* __restrict__ w,
                                  const float* __restrict__ bias,
                                  float* __restrict__ h) {
    int m0 = blockIdx.x * 16;
    int n0 = (threadIdx.x >> 5) * 16;
    v8f acc = {};
    acc = wmma_gemm_tile(h1, 64, m0, w, 32, n0, 64, acc);
    wmma_store_tile(h, 32, m0, n0, acc, bias, true);
}

// Layer 2: h2[N,16] = agg2[N,32]@w_rel2 + h[N,32]@w_root2 + b_rel2
// 8 waves/block, each wave one M-tile (wave-uniform guard keeps EXEC full).
__global__ void gcn_layer2_wmma(const float* __restrict__ agg2,
                                const float* __restrict__ h,
                                const float* __restrict__ wrel,
                                const float* __restrict__ wroot,
                                const float* __restrict__ bias,
                                float* __restrict__ h2) {
    int wave = threadIdx.x >> 5;
    int m0 = (blockIdx.x * 8 + wave) * 16;
    if (m0 >= NNODES) return;
    v8f acc = {};
    acc = wmma_gemm_tile(agg2, 32, m0, wrel, 16, 0, 32, acc);
    acc = wmma_gemm_tile(h,   32, m0, wroot, 16, 0, 32, acc);
    wmma_store_tile(h2, 16, m0, 0, acc, bias, false);
}

// out[N,10] = log_softmax(h2[N,16] @ w_l2 + b_l2); one thread per node.
__global__ void head_logsoftmax(const float* __restrict__ h2,
                                const float* __restrict__ w,
                                const float* __restrict__ b,
                                float* __restrict__ out) {
    int i = blockIdx.x * blockDim.x + threadIdx.x;
    if (i >= NNODES) return;
    float hv[16];
    #pragma unroll
    for (int k = 0; k < 16; ++k) hv[k] = h2[(long long)i * 16 + k];
    float logits[10];
    float mx = -1e30f;
    #pragma unroll
    for (int j = 0; j < 10; ++j) {
        float s = b[j];
        #pragma unroll
        for (int k = 0; k < 16; ++k) s = fmaf(hv[k], w[k * 10 + j], s);
        logits[j] = s;
        mx = fmaxf(mx, s);
    }
    float sum = 0.0f;
    #pragma unroll
    for (int j = 0; j < 10; ++j) sum += __expf(logits[j] - mx);
    float lse = mx + __logf(sum);
    #pragma unroll
    for (int j = 0; j < 10; ++j) out[(long long)i * 10 + j] = logits[j] - lse;
}

// ---------------------------------------------------------------- launcher

extern "C" void kernel_launch(void* const* d_in, const int* in_sizes, int n_in,
                              void* d_out, int out_size, void* d_ws, size_t ws_size,
                              hipStream_t stream) {
    const float* x      = (const float*)d_in[0];
    const int*   ei     = (const int*)d_in[1];     // [2, E] flat: src then dst
    const float* wrel1  = (const float*)d_in[2];
    const float* brel1  = (const float*)d_in[3];
    const float* wroot1 = (const float*)d_in[4];
    const float* wl1    = (const float*)d_in[5];
    const float* bl1    = (const float*)d_in[6];
    const float* wrel2  = (const float*)d_in[7];
    const float* brel2  = (const float*)d_in[8];
    const float* wroot2 = (const float*)d_in[9];
    const float* wl2    = (const float*)d_in[10];
    const float* bl2    = (const float*)d_in[11];
    float* out = (float*)d_out;

    const int* src = ei;
    const int* dst = ei + NEDGES;

    // Workspace layout (floats)
    float* ws   = (float*)d_ws;
    float* agg1 = ws;                                   // 20000*128
    float* h1   = agg1 + (size_t)NNODES * 128;          // 20000*64
    float* h    = h1   + (size_t)NNODES * 64;           // 20000*32
    float* agg2 = h    + (size_t)NNODES * 32;           // 20000*32
    float* h2   = agg2 + (size_t)NNODES * 32;           // 20000*16

    // 1) zero agg1, agg2
    {
        int n1 = NNODES * 128;
        zero_f32_kernel<<<(n1 + 255) / 256, 256, 0, stream>>>(agg1, n1);
        int n2 = NNODES * 32;
        zero_f32_kernel<<<(n2 + 255) / 256, 256, 0, stream>>>(agg2, n2);
    }

    // 2) layer-1 aggregation: agg1[dst] += x[src]   (E*32 threads)
    {
        long long threads = (long long)NEDGES * 32;
        int blocks = (int)((threads + 255) / 256);
        scatter_add_128<<<blocks, 256, 0, stream>>>(x, src, dst, agg1);
    }

    // 3) h1 = agg1@wrel1 + x@wroot1 + b1   (WMMA f32)
    gcn_layer1_wmma<<<NNODES / 16, 128, 0, stream>>>(agg1, x, wrel1, wroot1, brel1, h1);

    // 4) h = relu(h1@wl1 + bl1)
    linear1_relu_wmma<<<NNODES / 16, 64, 0, stream>>>(h1, wl1, bl1, h);

    // 5) layer-2 aggregation: agg2[dst] += h[src]
    {
        long long threads = (long long)NEDGES * 32;
        int blocks = (int)((threads + 255) / 256);
        scatter_add_32<<<blocks, 256, 0, stream>>>(h, src, dst, agg2);
    }

    // 6) h2 = agg2@wrel2 + h@wroot2 + b2
    {
        int mtiles = NNODES / 16;                 // 1250
        int blocks = (mtiles + 7) / 8;            // 8 waves per block
        gcn_layer2_wmma<<<blocks, 256, 0, stream>>>(agg2, h, wrel2, wroot2, brel2, h2);
    }

    // 7) out = log_softmax(h2@wl2 + bl2)
    head_logsoftmax<<<(NNODES + 255) / 256, 256, 0, stream>>>(h2, wl2, bl2, out);
}